// GCNHAConv_25778393711287
// MI455X (gfx1250) — compile-verified
//
#include <hip/hip_runtime.h>
#include <hip/hip_bf16.h>

typedef __attribute__((ext_vector_type(2))) float v2f;
typedef __attribute__((ext_vector_type(8))) float v8f;

#define NEG_SLOPE 0.2f

// ---------------------------------------------------------------- utilities
__global__ __launch_bounds__(256) void zero_kernel(float* __restrict__ p, long long count) {
    long long i = (long long)blockIdx.x * blockDim.x + threadIdx.x;
    if (i < count) p[i] = 0.0f;
}

__global__ __launch_bounds__(256) void deg_kernel(const int* __restrict__ dst,
                                                  float* __restrict__ deg, int E) {
    int e = blockIdx.x * blockDim.x + threadIdx.x;
    if (e < E) atomicAdd(deg + dst[e], 1.0f);
}

__global__ __launch_bounds__(256) void norm_kernel(float* __restrict__ norm, int N) {
    int n = blockIdx.x * blockDim.x + threadIdx.x;
    if (n < N) {
        float d = norm[n];
        norm[n] = rsqrtf(fmaxf(d, 1.0f));
    }
}

// out[n][:] = in[n][:] * norm[n]   (float4 per thread, 32 threads per row)
__global__ __launch_bounds__(256) void rowscale_kernel(float* __restrict__ out,
                                                       const float* __restrict__ in,
                                                       const float* __restrict__ norm,
                                                       long long total4) {
    long long tid = (long long)blockIdx.x * blockDim.x + threadIdx.x;
    if (tid >= total4) return;
    int n = (int)(tid >> 5);
    int c = ((int)tid & 31) * 4;
    float s = norm[n];
    float4 v = *(const float4*)(in + (long long)n * 128 + c);
    float4 r;
    r.x = v.x * s; r.y = v.y * s; r.z = v.z * s; r.w = v.w * s;
    *(float4*)(out + (long long)n * 128 + c) = r;
}

// ---------------------------------------------------------------- SpMM (scatter-add)
// One wave per edge (tid>>5 uniform within wave), one float4 per lane:
// 512B coalesced gather from scaled[src], 4 atomic f32 adds into agg[dst].
__global__ __launch_bounds__(256) void spmm_kernel(const float* __restrict__ scaled,
                                                   const int* __restrict__ src,
                                                   const int* __restrict__ dst,
                                                   float* __restrict__ agg,
                                                   long long total) {
    long long tid = (long long)blockIdx.x * blockDim.x + threadIdx.x;
    if (tid >= total) return;
    int e = (int)(tid >> 5);
    int c = ((int)tid & 31) * 4;
    int s = src[e];
    int d = dst[e];
    float4 v = *(const float4*)(scaled + (long long)s * 128 + c);
    float* o = agg + (long long)d * 128 + c;
    atomicAdd(o + 0, v.x);
    atomicAdd(o + 1, v.y);
    atomicAdd(o + 2, v.z);
    atomicAdd(o + 3, v.w);
}

// ---------------------------------------------------------------- WMMA GEMM
// proj[n][o] = sum_d h[n][d] * w[o][d]     (w row-major [OUT=128, D=128])
// Full-precision f32 path: V_WMMA_F32_16X16X4_F32, 32 K-steps of K=4.
// Block: 256 threads = 8 waves; wave w owns OUT-cols [16w,16w+16); block owns
// 32 rows (two 16x16 C tiles per wave to pipeline the WMMA RAW chain).
//
// ISA layouts (wave32):
//   A 16x4 f32 : lanes 0-15 -> K=k0,k0+1 (v0,v1) of row M=lane;
//                lanes 16-31 -> K=k0+2,k0+3 of row M=lane-16.
//   B 4x16 f32 : v0/v1 hold K=kh / kh+1 for col N=lane%16, kh = 2*(lane/16).
//   C 16x16 f32: VGPR i -> (M = i + 8*(lane/16), N = lane%16).
__global__ __launch_bounds__(256) void gemm_wmma_kernel(const float* __restrict__ h,
                                                        const float* __restrict__ w,
                                                        float* __restrict__ proj,
                                                        int nrows) {
    const int lane = threadIdx.x & 31;
    const int wave = threadIdx.x >> 5;     // 0..7 -> OUT col tile
    const int n0   = wave * 16;
    const int m0   = blockIdx.x * 32;

    const int lr = lane & 15;
    const int kh = (lane >> 4) * 2;        // 0 or 2

    int rowA = m0 + lr;
    int rowB = m0 + 16 + lr;
    if (rowA >= nrows) rowA = nrows - 1;   // clamp loads; stores are guarded
    if (rowB >= nrows) rowB = nrows - 1;

    const float* __restrict__ pa = h + (long long)rowA * 128 + kh;
    const float* __restrict__ pb = h + (long long)rowB * 128 + kh;
    const float* __restrict__ pw = w + (long long)(n0 + lr) * 128 + kh;

    v8f c0 = {};
    v8f c1 = {};
#pragma unroll
    for (int k0 = 0; k0 < 128; k0 += 4) {
        v2f a0 = *(const v2f*)(pa + k0);
        v2f a1 = *(const v2f*)(pb + k0);
        v2f b  = *(const v2f*)(pw + k0);
        c0 = __builtin_amdgcn_wmma_f32_16x16x4_f32(false, a0, false, b,
                                                   (short)0, c0, false, false);
        c1 = __builtin_amdgcn_wmma_f32_16x16x4_f32(false, a1, false, b,
                                                   (short)0, c1, false, false);
    }

    const int crow0 = m0 + ((lane >> 4) * 8);
    const int ccol  = n0 + lr;
#pragma unroll
    for (int i = 0; i < 8; ++i) {
        int r0 = crow0 + i;
        int r1 = r0 + 16;
        if (r0 < nrows) proj[(long long)r0 * 128 + ccol] = c0[i];
        if (r1 < nrows) proj[(long long)r1 * 128 + ccol] = c1[i];
    }
}

// ---------------------------------------------------------------- hop attention + combine
// One 128-thread block per node. out initially holds proj3; overwritten in place.
__global__ __launch_bounds__(128) void attn_combine_kernel(const float* __restrict__ p0,
                                                           const float* __restrict__ p1,
                                                           const float* __restrict__ p2,
                                                           const float* __restrict__ attn_l,
                                                           const float* __restrict__ attn_r,
                                                           float* __restrict__ out,
                                                           int N) {
    int n = blockIdx.x;
    int t = threadIdx.x;
    long long base = (long long)n * 128 + t;

    float v0 = p0[base];
    float v1 = p1[base];
    float v2 = p2[base];
    float v3 = out[base];          // proj3
    float al = attn_l[t];
    float ar = attn_r[t];

    float s[5] = { v0 * al, v0 * ar, v1 * ar, v2 * ar, v3 * ar };

    // wave32 tree reduce, then across the 4 waves through LDS
#pragma unroll
    for (int off = 16; off > 0; off >>= 1) {
#pragma unroll
        for (int i = 0; i < 5; ++i) s[i] += __shfl_down(s[i], off, 32);
    }

    __shared__ float red[5][4];
    __shared__ float wts[4];
    int wv = t >> 5;
    if ((t & 31) == 0) {
#pragma unroll
        for (int i = 0; i < 5; ++i) red[i][wv] = s[i];
    }
    __syncthreads();

    if (t == 0) {
        float tot[5];
#pragma unroll
        for (int i = 0; i < 5; ++i)
            tot[i] = red[i][0] + red[i][1] + red[i][2] + red[i][3];
        float sc[4];
        float m = -3.0e38f;
#pragma unroll
        for (int k = 0; k < 4; ++k) {
            float x = tot[0] + tot[k + 1];
            sc[k] = (x > 0.0f) ? x : NEG_SLOPE * x;   // leaky_relu
            m = fmaxf(m, sc[k]);
        }
        float e[4];
        float sum = 0.0f;
#pragma unroll
        for (int k = 0; k < 4; ++k) { e[k] = expf(sc[k] - m); sum += e[k]; }
        float inv = 1.0f / sum;
#pragma unroll
        for (int k = 0; k < 4; ++k) wts[k] = e[k] * inv;
    }
    __syncthreads();

    out[base] = wts[0] * v0 + wts[1] * v1 + wts[2] * v2 + wts[3] * v3;
}

// ---------------------------------------------------------------- launcher
extern "C" void kernel_launch(void* const* d_in, const int* in_sizes, int n_in,
                              void* d_out, int out_size, void* d_ws, size_t ws_size,
                              hipStream_t stream) {
    const float* feat   = (const float*)d_in[0];   // [N,128]
    const float* fc_w   = (const float*)d_in[1];   // [128,128]
    const float* attn_l = (const float*)d_in[2];   // [128]
    const float* attn_r = (const float*)d_in[3];   // [128]
    const int*   esrc   = (const int*)d_in[4];     // [E]
    const int*   edst   = (const int*)d_in[5];     // [E]

    const int N = in_sizes[0] / 128;
    const int E = in_sizes[4];
    float* out = (float*)d_out;

    // workspace layout (floats)
    float* ws     = (float*)d_ws;
    float* norm   = ws;                            // N        (doubles as deg)
    float* hbuf   = norm + N;                      // N*128    (ping h buffer)
    float* scaled = hbuf + (size_t)N * 128;        // N*128
    float* proj0  = scaled + (size_t)N * 128;      // N*128
    float* proj1  = proj0 + (size_t)N * 128;       // N*128
    float* proj2  = proj1 + (size_t)N * 128;       // N*128
    // proj3 lives in d_out and is combined in place.

    const int T = 256;
    const long long nf     = (long long)N * 128;
    const long long total4 = nf / 4;               // N*32 float4 workers
    const long long etotal = (long long)E * 32;    // E edges * 32 float4 lanes
    const int gemmBlocks   = (N + 31) / 32;

    // degree -> norm = clip(deg,1)^-0.5
    zero_kernel<<<(int)((N + T - 1) / T), T, 0, stream>>>(norm, N);
    deg_kernel<<<(E + T - 1) / T, T, 0, stream>>>(edst, norm, E);
    norm_kernel<<<(N + T - 1) / T, T, 0, stream>>>(norm, N);

    // proj0 = feat @ W^T
    gemm_wmma_kernel<<<gemmBlocks, 256, 0, stream>>>(feat, fc_w, proj0, N);

    // K=3 propagation hops; single ping h buffer (old h dead once `scaled` is built)
    float* projs[3] = { proj1, proj2, out };
    const float* hk = feat;
    for (int hop = 0; hop < 3; ++hop) {
        rowscale_kernel<<<(int)((total4 + T - 1) / T), T, 0, stream>>>(scaled, hk, norm, total4);
        zero_kernel<<<(int)((nf + T - 1) / T), T, 0, stream>>>(hbuf, nf);
        spmm_kernel<<<(int)((etotal + T - 1) / T), T, 0, stream>>>(scaled, esrc, edst, hbuf, etotal);
        rowscale_kernel<<<(int)((total4 + T - 1) / T), T, 0, stream>>>(hbuf, hbuf, norm, total4);
        gemm_wmma_kernel<<<gemmBlocks, 256, 0, stream>>>(hbuf, fc_w, projs[hop], N);
        hk = hbuf;
    }

    // hop softmax + weighted combine (overwrites proj3 in d_out)
    attn_combine_kernel<<<N, 128, 0, stream>>>(proj0, proj1, proj2, attn_l, attn_r, out, N);
}